// custom_weighted_loss_3_10514079940985
// MI455X (gfx1250) — compile-verified
//
#include <hip/hip_runtime.h>
#include <hip/hip_bf16.h>

// Problem shape (compile-time constants from the reference)
constexpr int B_ = 16, C_ = 13, H_ = 512, W_ = 512;
constexpr int HW  = H_ * W_;        // 262144 pixels per (b,c) plane
constexpr int HW4 = HW / 4;         // 65536 float4 groups per plane
constexpr long long NTOT = (long long)B_ * C_ * H_ * W_;  // 54,525,952

typedef __attribute__((ext_vector_type(2))) float v2f;
typedef __attribute__((ext_vector_type(4))) float f4;
typedef __attribute__((ext_vector_type(8))) float v8f;

// ---- CDNA5 WMMA f32 16x16x4 wave32 reduction -------------------------------
// A[m,0] = x(lane m), A[m,2] = x(lane m+16), A[m,1]=A[m,3]=0, B = ones:
//   D[m][n] = x[m] + x[m+16]  (independent of n)
// D layout: vgpr r holds row r on lanes 0-15 / row r+8 on lanes 16-31, so
// sum(d[0..7]) yields half the wave total on lanes 0-15 and the other half on
// lanes 16-31; one xor-16 shuffle completes the sum on every lane.
// EXEC must be all ones at the call site.
__device__ __forceinline__ float wave_reduce_wmma(float x) {
  v2f a; a[0] = x;    a[1] = 0.0f;
  v2f b; b[0] = 1.0f; b[1] = 1.0f;
  v8f c = {};
  v8f d = __builtin_amdgcn_wmma_f32_16x16x4_f32(
      /*neg_a=*/false, a, /*neg_b=*/false, b,
      /*c_mod=*/(short)0, c, /*reuse_a=*/false, /*reuse_b=*/false);
  float s = ((d[0] + d[1]) + (d[2] + d[3])) + ((d[4] + d[5]) + (d[6] + d[7]));
  s += __shfl_xor(s, 16, 32);
  return s;  // full 32-lane sum on every lane
}

// ---- Kernel 0: zero the workspace words we use -----------------------------
__global__ void __launch_bounds__(256) k_zero(int* __restrict__ ws, int nwords) {
  for (int i = threadIdx.x; i < nwords; i += blockDim.x) ws[i] = 0;
}

// ---- Kernel 1: active[b,c] = any(target > 0.5) over the plane --------------
// Default (RT) temporal hint: we WANT target resident in L2 for pass 2.
__global__ void __launch_bounds__(256) k_active(const float* __restrict__ tgt,
                                                int* __restrict__ flags) {
  const int plane = blockIdx.y;                 // b*C + c, 0..207
  const f4* t4 = reinterpret_cast<const f4*>(tgt) + (size_t)plane * HW4;
  const int tid    = blockIdx.x * blockDim.x + threadIdx.x;
  const int stride = gridDim.x * blockDim.x;

  bool any = false;
  for (int i = tid; i < HW4; i += stride) {
    const f4 t = t4[i];
    any = any || (t.x > 0.5f) || (t.y > 0.5f) || (t.z > 0.5f) || (t.w > 0.5f);
  }

  __shared__ int sflag;
  if (threadIdx.x == 0) sflag = 0;
  __syncthreads();
  if (__any(any) && (threadIdx.x & 31) == 0) sflag = 1;  // benign race: all write 1
  __syncthreads();
  if (threadIdx.x == 0 && sflag) atomicOr(&flags[plane], 1);
}

// ---- Kernel 2: fused weighted-diff2 + hierarchical reduction ---------------
// launch_bounds(256, 1): memory-bound kernel with 26-deep per-thread MLP;
// trade occupancy for registers so the fully unrolled body doesn't spill.
__global__ void __launch_bounds__(256, 1) k_loss(const float* __restrict__ pred,
                                                 const float* __restrict__ tgt,
                                                 const int* __restrict__ flags,
                                                 float* __restrict__ partials) {
  const int b = blockIdx.y;

  __shared__ unsigned smask;
  __shared__ float wsum[8];
  if (threadIdx.x == 0) {
    unsigned m = 0;
    int act[C_];
    for (int c = 0; c < C_; ++c) act[c] = flags[b * C_ + c];
    for (int c = 0; c < C_; ++c) {
      int mm = act[c];
      if (c > 0)      mm |= act[c - 1];
      if (c < C_ - 1) mm |= act[c + 1];
      if (mm) m |= (1u << c);
    }
    smask = m;
  }
  __syncthreads();
  // Force the mask into an SGPR so per-channel selects are scalar.
  const unsigned mask = __builtin_amdgcn_readfirstlane(smask);

  const size_t base4 = (size_t)b * C_ * HW4;
  const f4* t4 = reinterpret_cast<const f4*>(tgt)  + base4;
  const f4* p4 = reinterpret_cast<const f4*>(pred) + base4;

  const int tid    = blockIdx.x * blockDim.x + threadIdx.x;
  const int stride = gridDim.x * blockDim.x;

  float acc = 0.0f;
  for (int i = tid; i < HW4; i += stride) {
    float we0 = 0.f, we1 = 0.f, we2 = 0.f, we3 = 0.f;   // channel counts
    float sa0 = 0.f, sa1 = 0.f, sa2 = 0.f, sa3 = 0.f;   // all-channel diff2
    float sm0 = 0.f, sm1 = 0.f, sm2 = 0.f, sm3 = 0.f;   // masked-channel diff2
#pragma unroll
    for (int c = 0; c < C_; ++c) {
      const float fm = (mask & (1u << c)) ? 1.0f : 0.0f;  // uniform scalar
      const f4 t = t4[(size_t)c * HW4 + i];
      // prediction is single-use: non-temporal so it doesn't evict target in L2
      const f4 p = __builtin_nontemporal_load(&p4[(size_t)c * HW4 + i]);
      float d0 = p.x - t.x, d1 = p.y - t.y, d2 = p.z - t.z, d3 = p.w - t.w;
      d0 *= d0; d1 *= d1; d2 *= d2; d3 *= d3;
      we0 += (t.x > 0.5f) ? 1.0f : 0.0f;
      we1 += (t.y > 0.5f) ? 1.0f : 0.0f;
      we2 += (t.z > 0.5f) ? 1.0f : 0.0f;
      we3 += (t.w > 0.5f) ? 1.0f : 0.0f;
      sa0 += d0; sa1 += d1; sa2 += d2; sa3 += d3;        // unconditional
      sm0 += d0 * fm; sm1 += d1 * fm; sm2 += d2 * fm; sm3 += d3 * fm;  // fma
    }
    // (we*9+1)*masked + 5*(we+1)*(all - masked)
    acc += (we0 * 9.f + 1.f) * sm0 + (we0 + 1.f) * 5.f * (sa0 - sm0);
    acc += (we1 * 9.f + 1.f) * sm1 + (we1 + 1.f) * 5.f * (sa1 - sm1);
    acc += (we2 * 9.f + 1.f) * sm2 + (we2 + 1.f) * 5.f * (sa2 - sm2);
    acc += (we3 * 9.f + 1.f) * sm3 + (we3 + 1.f) * 5.f * (sa3 - sm3);
  }

  // Wave reduction via WMMA (all 256 threads active -> EXEC all ones)
  const float wtot = wave_reduce_wmma(acc);
  const int warp = threadIdx.x >> 5;
  if ((threadIdx.x & 31) == 0) wsum[warp] = wtot;
  __syncthreads();
  if (threadIdx.x == 0) {
    float s = 0.f;
    for (int k = 0; k < 8; ++k) s += wsum[k];
    partials[blockIdx.y * gridDim.x + blockIdx.x] = s;
  }
}

// ---- Kernel 3: final deterministic reduction of block partials -------------
__global__ void __launch_bounds__(256) k_final(const float* __restrict__ partials,
                                               float* __restrict__ out) {
  constexpr int NP = 1024;  // 64 x 16 blocks from k_loss
  __shared__ float wsum[8];
  float acc = 0.0f;
#pragma unroll
  for (int k = 0; k < NP / 256; ++k) acc += partials[threadIdx.x + 256 * k];

  const float wtot = wave_reduce_wmma(acc);  // EXEC all ones, 256 threads
  const int warp = threadIdx.x >> 5;
  if ((threadIdx.x & 31) == 0) wsum[warp] = wtot;
  __syncthreads();
  if (threadIdx.x == 0) {
    double s = 0.0;
    for (int k = 0; k < 8; ++k) s += (double)wsum[k];
    out[0] = (float)(s / (double)NTOT);
  }
}

// ---- Host-side launch ------------------------------------------------------
extern "C" void kernel_launch(void* const* d_in, const int* in_sizes, int n_in,
                              void* d_out, int out_size, void* d_ws, size_t ws_size,
                              hipStream_t stream) {
  const float* pred = (const float*)d_in[0];  // prediction
  const float* tgt  = (const float*)d_in[1];  // target
  float* out = (float*)d_out;

  // Workspace layout: words [0,208) int flags ; words [256, 256+1024) float partials
  int*   flags    = (int*)d_ws;
  float* partials = (float*)((char*)d_ws + 1024);

  k_zero  <<<1, 256, 0, stream>>>(flags, 256 + 1024);
  k_active<<<dim3(32, B_ * C_), 256, 0, stream>>>(tgt, flags);
  k_loss  <<<dim3(64, B_),      256, 0, stream>>>(pred, tgt, flags, partials);
  k_final <<<1, 256, 0, stream>>>(partials, out);
}